// CenterLoss_89232240542211
// MI455X (gfx1250) — compile-verified
//
#include <hip/hip_runtime.h>

#define BB   8192
#define NC   1000
#define DD   2048
#define BM   128
#define BN   128
#define KC   32
#define LDA  36   // padded LDS row stride (floats): 144B rows -> 16B aligned, conflict-free

typedef __attribute__((ext_vector_type(2))) float v2f;
typedef __attribute__((ext_vector_type(4))) float v4f;
typedef __attribute__((ext_vector_type(8))) float v8f;

__global__ void cl_init_ws(double* acc, float* out) {
    if (threadIdx.x < 2) acc[threadIdx.x] = 0.0;
    if (threadIdx.x == 0) out[0] = 0.0f;
}

// One wave (32 lanes) per row: ||x_b||^2 into xx[], ||c_j||^2 into cc[]
__global__ __launch_bounds__(256)
void cl_row_norms(const float* __restrict__ x, const float* __restrict__ cen,
                  float* __restrict__ xx, float* __restrict__ cc) {
    const int wave = (int)(blockIdx.x * (blockDim.x >> 5)) + ((int)threadIdx.x >> 5);
    const int lane = threadIdx.x & 31;
    const int nrows = BB + NC;
    if (wave >= nrows) return;
    const float* src = (wave < BB) ? (x + (size_t)wave * DD)
                                   : (cen + (size_t)(wave - BB) * DD);
    float s = 0.f;
    for (int i = lane; i < DD / 4; i += 32) {
        v4f v = *(const v4f*)(src + i * 4);
        s += v.x * v.x + v.y * v.y + v.z * v.z + v.w * v.w;
    }
    for (int off = 16; off > 0; off >>= 1) s += __shfl_xor(s, off, 32);
    if (lane == 0) {
        if (wave < BB) xx[wave] = s;
        else           cc[wave - BB] = s;
    }
}

// 256 threads = 8 waves (2 wave-rows x 4 wave-cols). Block tile 128x128;
// each wave computes a 4x2 grid of 16x16 WMMA subtiles (8 independent
// accumulator chains -> XDL ILP, 6 LDS frag loads feed 8 WMMAs).
__global__ __launch_bounds__(256)
void cl_dist_tiles(const float* __restrict__ x, const float* __restrict__ cen,
                   const int* __restrict__ labels,
                   const float* __restrict__ xx, const float* __restrict__ ccn,
                   double* __restrict__ acc) {
    __shared__ __align__(16) float As[BM * LDA];
    __shared__ __align__(16) float Bs[BN * LDA];
    __shared__ double redT[256];
    __shared__ double redM[256];

    const int tid  = threadIdx.x;
    const int lane = tid & 31;
    const int wave = tid >> 5;
    const int wm   = wave & 1;    // wave row: 0..1 (64 rows each)
    const int wn   = wave >> 1;   // wave col: 0..3 (32 cols each)
    const int bm0  = blockIdx.x * BM;
    const int bn0  = blockIdx.y * BN;

    // fragment lane mapping (32-bit 16x4 A / 4x16 B layout)
    const int fm = lane & 15;     // M (A) or N (B)
    const int fh = lane >> 4;     // half: K offset of +2

    v8f acc_d[4][2] = {};         // 8 independent 16x16 accumulators

    // cooperative staging: thread t -> row t/2, 16 contiguous floats
    const int row  = tid >> 1;            // 0..127
    const int col0 = (tid & 1) * 16;      // 0 or 16
    const bool bvalid = (bn0 + row) < NC;

    const float* xg = x   + (size_t)(bm0 + row) * DD + col0;
    const float* cg = cen + (size_t)(bn0 + row) * DD + col0;

    for (int k0 = 0; k0 < DD; k0 += KC) {
#pragma unroll
        for (int q = 0; q < 4; ++q)
            *(v4f*)&As[row * LDA + col0 + q * 4] = *(const v4f*)(xg + k0 + q * 4);
        if (bvalid) {
#pragma unroll
            for (int q = 0; q < 4; ++q)
                *(v4f*)&Bs[row * LDA + col0 + q * 4] = *(const v4f*)(cg + k0 + q * 4);
        } else {
            v4f z = {0.f, 0.f, 0.f, 0.f};
#pragma unroll
            for (int q = 0; q < 4; ++q)
                *(v4f*)&Bs[row * LDA + col0 + q * 4] = z;
        }
        __syncthreads();
#pragma unroll
        for (int kk = 0; kk < KC; kk += 4) {
            v2f a[4], b[2];
#pragma unroll
            for (int i = 0; i < 4; ++i)
                a[i] = *(const v2f*)&As[(wm * 64 + i * 16 + fm) * LDA + kk + 2 * fh];
#pragma unroll
            for (int j = 0; j < 2; ++j)
                b[j] = *(const v2f*)&Bs[(wn * 32 + j * 16 + fm) * LDA + kk + 2 * fh];
#pragma unroll
            for (int i = 0; i < 4; ++i)
#pragma unroll
                for (int j = 0; j < 2; ++j)
                    acc_d[i][j] = __builtin_amdgcn_wmma_f32_16x16x4_f32(
                        false, a[i], false, b[j], (short)0, acc_d[i][j], false, false);
        }
        __syncthreads();
    }

    // epilogue: d = ||x||^2 + ||c||^2 - 2*dot, clamp, accumulate
    int   gncol[2]; float cnv[2]; bool nval[2];
#pragma unroll
    for (int j = 0; j < 2; ++j) {
        gncol[j] = bn0 + wn * 32 + j * 16 + fm;
        nval[j]  = gncol[j] < NC;
        cnv[j]   = nval[j] ? ccn[gncol[j]] : 0.f;
    }
    float ts = 0.f, ms = 0.f;
#pragma unroll
    for (int i = 0; i < 4; ++i) {
#pragma unroll
        for (int jj = 0; jj < 8; ++jj) {
            const int gm = bm0 + wm * 64 + i * 16 + jj + 8 * fh;  // C/D layout: M=jj (+8 hi lanes)
            const float xv = xx[gm];
            const int   lb = labels[gm];
#pragma unroll
            for (int j = 0; j < 2; ++j) {
                if (nval[j]) {
                    float d = xv + cnv[j] - 2.0f * acc_d[i][j][jj];
                    d = fminf(fmaxf(d, 1e-12f), 1e12f);
                    ts += d;
                    if (lb == gncol[j]) ms += d;
                }
            }
        }
    }
    redT[tid] = (double)ts; redM[tid] = (double)ms;
    __syncthreads();
    for (int s = 128; s > 0; s >>= 1) {
        if (tid < s) { redT[tid] += redT[tid + s]; redM[tid] += redM[tid + s]; }
        __syncthreads();
    }
    if (tid == 0) {
        atomicAdd(&acc[0], redT[0]);
        atomicAdd(&acc[1], redM[0]);
    }
}

__global__ void cl_finalize(const double* acc, float* out) {
    if (threadIdx.x == 0) {
        double total  = acc[0];
        double masked = acc[1];
        double center = masked / (double)BB;
        double sep    = (total - masked) / ((double)BB * (double)(NC - 1));
        out[0] = (float)(center - 0.001 * sep);
    }
}

extern "C" void kernel_launch(void* const* d_in, const int* in_sizes, int n_in,
                              void* d_out, int out_size, void* d_ws, size_t ws_size,
                              hipStream_t stream) {
    const float* x      = (const float*)d_in[0];
    const float* cen    = (const float*)d_in[1];
    const int*   labels = (const int*)d_in[2];
    float*  out = (float*)d_out;
    double* acc = (double*)d_ws;                       // [2] f64 accumulators
    float*  xx  = (float*)((char*)d_ws + 16);          // [BB] row norms of x
    float*  ccn = xx + BB;                             // [NC] row norms of centers

    cl_init_ws<<<1, 64, 0, stream>>>(acc, out);

    const int nrows = BB + NC;
    const int waves_per_block = 256 / 32;
    const int nb = (nrows + waves_per_block - 1) / waves_per_block;
    cl_row_norms<<<nb, 256, 0, stream>>>(x, cen, xx, ccn);

    dim3 grid(BB / BM, (NC + BN - 1) / BN);            // 64 x 8 (cols padded to 1024)
    cl_dist_tiles<<<grid, 256, 0, stream>>>(x, cen, labels, xx, ccn, acc);

    cl_finalize<<<1, 32, 0, stream>>>(acc, out);
}